// GCN_37580963840689
// MI455X (gfx1250) — compile-verified
//
#include <hip/hip_runtime.h>

// ---------------------------------------------------------------------------
// GCN forward on MI455X (gfx1250, wave32, WMMA + TDM).
//   h1 = relu(spmm(x @ W1) + b1)
//   h2 = relu(spmm(h1 @ W2) + b2)
//   out = [h2 @ WL1 + bL1 ; h2 @ WL2 + bL2]
// GEMMs: V_WMMA_F32_16X16X4_F32 (true fp32 matrix pipe). A panels staged in
// LDS by the Tensor Data Mover (tensor_load_to_lds + s_wait_tensorcnt), with
// TDM LDS padding to avoid bank conflicts. SpMM: sorted edge_dst -> per-node
// binary-searched segment, float4 (b128) coalesced gathers, deterministic.
// ---------------------------------------------------------------------------

typedef __attribute__((ext_vector_type(2))) float v2f;
typedef __attribute__((ext_vector_type(8))) float v8f;
typedef __attribute__((ext_vector_type(4))) unsigned int u32x4;
typedef __attribute__((ext_vector_type(8))) int i32x8;
typedef __attribute__((ext_vector_type(4))) int i32x4;

#define KDIM 256   // inner dim is always 256 in this model
#define LDS_PAD 4  // 4-dword pad per 256-dword row (TDM pad_interval=7, pad_amount=3)

// C[M x N] = op(A[M x 256]) @ W[256 x N] (+ bias), op = optional ReLU.
// Block: 256 threads = 8 waves, 16 output tiles (2 per wave, sharing one A
// fragment). N=256 -> block covers 16 rows; N=64 -> 64 rows.
template <int N, int RELU_A>
__global__ void gemm_wmma_f32(const float* __restrict__ A,
                              const float* __restrict__ W,
                              const float* __restrict__ bias,
                              float* __restrict__ C, int M) {
  constexpr int TN = N / 16;          // n-tiles per 16-row panel
  constexpr int PANELS = 16 / TN;     // row panels per block (16 tiles/block)
  constexpr int ROWS = 16 * PANELS;   // rows of A staged per block
  static_assert(TN >= 2 && 16 % TN == 0, "tile mapping");

  __shared__ float As[ROWS][KDIM + LDS_PAD];

  const int row0 = blockIdx.x * ROWS;

  // ---- Stage A panel (ROWS x 256 f32) into LDS ----
#if defined(__HIP_DEVICE_COMPILE__) && __has_builtin(__builtin_amdgcn_tensor_load_to_lds)
  if (threadIdx.x < 32) {  // one wave issues the TDM descriptor
    const unsigned long long ga =
        (unsigned long long)(uintptr_t)(A + (size_t)row0 * KDIM);
    const unsigned rows_rem = (unsigned)(M - row0);  // TDM zero-fills OOB rows
    u32x4 g0;
    g0[0] = 1u;                                   // count=1, user descriptor
    g0[1] = (unsigned)(uintptr_t)(&As[0][0]);     // LDS byte address (addr[31:0])
    g0[2] = (unsigned)(ga & 0xFFFFFFFFu);         // global_addr[31:0]
    g0[3] = (unsigned)((ga >> 32) & 0x01FFFFFFu)  // global_addr[56:32]
            | (2u << 30);                         // type=2 ("image")
    i32x8 g1;
    g1[0] = (int)((2u << 16)      // data_size = 4B
                  | (1u << 20)    // pad_enable
                  | (7u << 22)    // pad_interval: every 256 dwords
                  | (3u << 25));  // pad_amount: 4 dwords
    g1[1] = (int)(((unsigned)KDIM & 0xFFFFu) << 16);                 // tensor_dim0[15:0]
    g1[2] = (int)((((unsigned)KDIM >> 16) & 0xFFFFu)                 // tensor_dim0[31:16]
                  | ((rows_rem & 0xFFFFu) << 16));                   // tensor_dim1[15:0]
    g1[3] = (int)(((rows_rem >> 16) & 0xFFFFu)                       // tensor_dim1[31:16]
                  | ((unsigned)KDIM << 16));                         // tile_dim0 = 256
    g1[4] = (int)(unsigned)ROWS;   // tile_dim1 = ROWS, tile_dim2 = 0
    g1[5] = (int)(unsigned)KDIM;   // tensor_dim0_stride = 256 elements
    g1[6] = 0;
    g1[7] = 0;
    i32x4 gz4 = {0, 0, 0, 0};      // groups 2/3 unused (2D tile)
    i32x8 gz8 = {0, 0, 0, 0, 0, 0, 0, 0};
    __builtin_amdgcn_tensor_load_to_lds(g0, g1, gz4, gz4, gz8, 0);
    __builtin_amdgcn_s_wait_tensorcnt(0);
  }
  __syncthreads();
#else
  for (int i = threadIdx.x; i < ROWS * KDIM; i += 256) {
    const int r = i >> 8, c = i & (KDIM - 1);
    const int gr = row0 + r;
    As[r][c] = (gr < M) ? A[(size_t)gr * KDIM + c] : 0.f;
  }
  __syncthreads();
#endif

  // ---- Per-wave tile mapping: wave w owns tiles 2w, 2w+1 (same row panel) ----
  const int wv = threadIdx.x >> 5;
  const int lane = threadIdx.x & 31;
  const int t0 = wv << 1;
  const int panel = t0 / TN;       // uniform per wave (2 divides TN)
  const int tn0 = t0 % TN;
  const int half = lane >> 4;      // lane half selects K sub-pair
  const int l16 = lane & 15;
  const int col0 = (tn0 << 4) + l16;
  const int col1 = col0 + 16;
  const int kh = half << 1;

  const float* __restrict__ as_row = &As[(panel << 4) + l16][0];
  v8f acc0 = {}, acc1 = {};

#pragma unroll 4
  for (int k = 0; k < KDIM; k += 4) {
    v2f a = *(const v2f*)(as_row + k + kh);   // ds_load_b64
    if (RELU_A) {
      a.x = fmaxf(a.x, 0.f);
      a.y = fmaxf(a.y, 0.f);
    }
    v2f b0, b1;
    b0.x = W[(k + kh) * N + col0];
    b0.y = W[(k + kh + 1) * N + col0];
    b1.x = W[(k + kh) * N + col1];
    b1.y = W[(k + kh + 1) * N + col1];
    acc0 = __builtin_amdgcn_wmma_f32_16x16x4_f32(false, a, false, b0, (short)0,
                                                 acc0, false, false);
    acc1 = __builtin_amdgcn_wmma_f32_16x16x4_f32(false, a, false, b1, (short)0,
                                                 acc1, false, false);
  }

  // C/D layout: VGPR r -> M = panel*16 + half*8 + r, N = col
  const int browbase = row0 + (panel << 4) + (half << 3);
  const float bv0 = bias ? bias[col0] : 0.f;
  const float bv1 = bias ? bias[col1] : 0.f;
#pragma unroll
  for (int r = 0; r < 8; ++r) {
    const int grow = browbase + r;
    if (grow < M) {
      C[(size_t)grow * N + col0] = acc0[r] + bv0;
      C[(size_t)grow * N + col1] = acc1[r] + bv1;
    }
  }
}

// out[v, :] = bias + sum_{e : dst[e]==v} w[e] * h[src[e], :]
// 4 nodes per 256-thread block; 64 lanes x float4 per node row (b128 gathers,
// L2-resident: h is 102.4MB < 192MB L2). Deterministic in-edge-order sum.
__global__ void spmm_bias(const int* __restrict__ esrc,
                          const int* __restrict__ edst,
                          const float* __restrict__ ew,
                          const float* __restrict__ h,
                          const float* __restrict__ bias,
                          float* __restrict__ out, int nE, int M) {
  const int sub = threadIdx.x >> 6;        // node slot 0..3 (wave-uniform)
  const int fq = (threadIdx.x & 63) << 2;  // feature quad start
  const int v = (blockIdx.x << 2) + sub;
  if (v >= M) return;

  // lower/upper bound over sorted edst (uniform per wave -> scalarizes)
  int lo = 0, hi = nE;
  while (lo < hi) {
    const int mid = (lo + hi) >> 1;
    if (edst[mid] < v) lo = mid + 1; else hi = mid;
  }
  int e = lo;
  int lo2 = lo;
  hi = nE;
  while (lo2 < hi) {
    const int mid = (lo2 + hi) >> 1;
    if (edst[mid] <= v) lo2 = mid + 1; else hi = mid;
  }
  const int e_end = lo2;

  float a0 = bias[fq], a1 = bias[fq + 1], a2 = bias[fq + 2], a3 = bias[fq + 3];
  for (; e < e_end; ++e) {
    const float wgt = ew[e];
    const float4 hv = *(const float4*)(h + (size_t)esrc[e] * KDIM + fq);
    a0 = fmaf(wgt, hv.x, a0);
    a1 = fmaf(wgt, hv.y, a1);
    a2 = fmaf(wgt, hv.z, a2);
    a3 = fmaf(wgt, hv.w, a3);
  }
  float4 res;
  res.x = a0; res.y = a1; res.z = a2; res.w = a3;
  *(float4*)(out + (size_t)v * KDIM + fq) = res;
}

extern "C" void kernel_launch(void* const* d_in, const int* in_sizes, int n_in,
                              void* d_out, int out_size, void* d_ws,
                              size_t ws_size, hipStream_t stream) {
  const float* x    = (const float*)d_in[0];
  const int*   esrc = (const int*)d_in[1];
  const int*   edst = (const int*)d_in[2];
  const float* ew   = (const float*)d_in[3];
  const float* W1   = (const float*)d_in[4];
  const float* b1   = (const float*)d_in[5];
  const float* W2   = (const float*)d_in[6];
  const float* b2   = (const float*)d_in[7];
  const float* WL1  = (const float*)d_in[8];
  const float* bL1  = (const float*)d_in[9];
  const float* WL2  = (const float*)d_in[10];
  const float* bL2  = (const float*)d_in[11];
  float* out = (float*)d_out;

  const int M  = in_sizes[0] / KDIM;  // 100000 nodes
  const int nE = in_sizes[1];         // 3.2M edges

  float* buf0 = (float*)d_ws;             // M x 256 scratch (ping)
  float* buf1 = buf0 + (size_t)M * KDIM;  // M x 256 scratch (pong)

  const int gb256 = (M + 15) / 16;   // gemm<256>: 16 rows per block
  const int gb64  = (M + 63) / 64;   // gemm<64>:  64 rows per block
  const int sb    = (M + 3) / 4;     // spmm: 4 nodes per block

  // Layer 1
  gemm_wmma_f32<KDIM, 0><<<gb256, 256, 0, stream>>>(x, W1, nullptr, buf0, M);
  spmm_bias<<<sb, 256, 0, stream>>>(esrc, edst, ew, buf0, b1, buf1, nE, M);

  // Layer 2 (ReLU fused into A-read of the consuming GEMM)
  gemm_wmma_f32<KDIM, 1><<<gb256, 256, 0, stream>>>(buf1, W2, nullptr, buf0, M);
  spmm_bias<<<sb, 256, 0, stream>>>(esrc, edst, ew, buf0, b2, buf1, nE, M);

  // Twin heads
  gemm_wmma_f32<64, 1><<<gb64, 256, 0, stream>>>(buf1, WL1, bL1, out, M);
  gemm_wmma_f32<64, 1><<<gb64, 256, 0, stream>>>(buf1, WL2, bL2,
                                                 out + (size_t)M * 64, M);
}